// HGTLayer_72258529788471
// MI455X (gfx1250) — compile-verified
//
#include <hip/hip_runtime.h>
#include <hip/hip_bf16.h>

// HGT layer for MI455X (gfx1250, wave32).
// Dense phase: fp32 WMMA (v_wmma_f32_16x16x4_f32) GEMMs with relation
// transforms algebraically fused into the projection weights.
// Sparse phase: bandwidth-bound edge gather/softmax/scatter with f32 atomics.

typedef __attribute__((ext_vector_type(2))) float v2f;
typedef __attribute__((ext_vector_type(8))) float v8f;

#define IN_DIM 128
#define OUT_DIM 128
#define NHEAD 8
#define DKDIM 16

// ---------------------------------------------------------------------------
// Fuse per-head relation transform into projection weight:
//   Wf[o=h*16+j][c] = sum_i Wsrc[h*16+i][c] * rel[h][i][j]
//   bf[o=h*16+j]    = sum_i bsrc[h*16+i]    * rel[h][i][j]
// ---------------------------------------------------------------------------
__global__ __launch_bounds__(256) void fuse_weights(
    const float* __restrict__ Wsrc, const float* __restrict__ bsrc,
    const float* __restrict__ rel, float* __restrict__ Wf,
    float* __restrict__ bf) {
  int idx = blockIdx.x * blockDim.x + threadIdx.x;
  if (idx >= 128 * 128) return;
  int o = idx >> 7, c = idx & 127;
  int hh = o >> 4, j = o & 15;
  float s = 0.f;
#pragma unroll
  for (int i = 0; i < 16; ++i)
    s += Wsrc[(hh * 16 + i) * 128 + c] * rel[(hh * 16 + i) * 16 + j];
  Wf[o * 128 + c] = s;
  if (c == 0) {
    float b = 0.f;
#pragma unroll
    for (int i = 0; i < 16; ++i)
      b += bsrc[hh * 16 + i] * rel[(hh * 16 + i) * 16 + j];
    bf[o] = b;
  }
}

// ---------------------------------------------------------------------------
// C[M x 128] = A[M x 128] @ W.T + bias, optional skip-sigmoid residual blend:
//   C = (A@W.T + b) * sigmoid(skip) + resid * (1 - sigmoid(skip))
// W stored torch-style [o][c]. 256 threads = 8 wave32; wave w computes the
// 16x128 output slice for rows [blk*128 + w*16, +16) with 8 fp32 WMMA tiles.
// K is chunked by 32 through LDS (pad 36 floats -> conflict-free b64 reads).
// ---------------------------------------------------------------------------
__global__ __launch_bounds__(256) void gemm128_wmma(
    const float* __restrict__ A, const float* __restrict__ W,
    const float* __restrict__ bias, float* __restrict__ C,
    const float* __restrict__ resid, const float* __restrict__ skip_p,
    int M) {
  __shared__ float As[128][36];
  __shared__ float Wl[128][36];

  const int tid = threadIdx.x;
  const int wave = tid >> 5;
  const int lane = tid & 31;
  const int half = lane >> 4;  // 0: lanes 0-15, 1: lanes 16-31
  const int m16 = lane & 15;
  const int row_base = blockIdx.x * 128;

  v8f acc[8];
#pragma unroll
  for (int t = 0; t < 8; ++t) acc[t] = 0;

  for (int kc = 0; kc < 128; kc += 32) {
    // stage W chunk: Wl[o][c_local], coalesced float4 loads
#pragma unroll
    for (int i = 0; i < 4; ++i) {
      int idx = tid + i * 256;  // 128 o * 8 float4
      int o = idx >> 3, cc = idx & 7;
      float4 w = *(const float4*)(W + (size_t)o * 128 + kc + cc * 4);
      *(float4*)(&Wl[o][cc * 4]) = w;
    }
    // stage A chunk: As[row][c_local], zero-pad OOB rows
#pragma unroll
    for (int i = 0; i < 4; ++i) {
      int idx = tid + i * 256;
      int r = idx >> 3, cc = idx & 7;
      int g = row_base + r;
      float4 a = make_float4(0.f, 0.f, 0.f, 0.f);
      if (g < M) a = *(const float4*)(A + (size_t)g * 128 + kc + cc * 4);
      *(float4*)(&As[r][cc * 4]) = a;
    }
    __syncthreads();

    const float* arow = &As[wave * 16 + m16][0];
#pragma unroll
    for (int kk = 0; kk < 32; kk += 4) {
      // A fragment 16x4: VGPR0 = K{0,2}(per half), VGPR1 = K{1,3}
      float2 at = *(const float2*)(arow + kk + 2 * half);
      v2f va;
      va.x = at.x;
      va.y = at.y;
#pragma unroll
      for (int ot = 0; ot < 8; ++ot) {
        // B fragment 4x16: lane m holds column n=m; rows K{0,2}/{1,3} by half
        float2 bt = *(const float2*)(&Wl[ot * 16 + m16][kk + 2 * half]);
        v2f vb;
        vb.x = bt.x;
        vb.y = bt.y;
        acc[ot] = __builtin_amdgcn_wmma_f32_16x16x4_f32(
            false, va, false, vb, (short)0, acc[ot], false, false);
      }
    }
    __syncthreads();
  }

  float alpha = 1.f, beta = 0.f;
  if (skip_p) {
    float s = *skip_p;
    alpha = 1.f / (1.f + __expf(-s));
    beta = 1.f - alpha;
  }
#pragma unroll
  for (int ot = 0; ot < 8; ++ot) {
    int col = ot * 16 + m16;
    float b = bias ? bias[col] : 0.f;
#pragma unroll
    for (int r = 0; r < 8; ++r) {
      int row = row_base + wave * 16 + r + 8 * half;  // C layout: VGPR r -> M=r / M=8+r
      if (row < M) {
        float v = acc[ot][r] + b;
        if (resid) v = v * alpha + resid[(size_t)row * 128 + col] * beta;
        C[(size_t)row * 128 + col] = v;
      }
    }
  }
}

// ---------------------------------------------------------------------------
// Edge phase helpers
// ---------------------------------------------------------------------------
__device__ __forceinline__ void atomicMaxF(float* addr, float val) {
  if (!(val == val)) return;
  if (val >= 0.f)
    atomicMax((int*)addr, __float_as_int(val));
  else
    atomicMin((unsigned int*)addr, __float_as_uint(val));
}

// pass 1: per-(edge,head) score = <q[dst], kr[src]>_head * pri[h]/sqrt(dk); seg-max
__global__ __launch_bounds__(256) void edge_score_max(
    const int* __restrict__ src, const int* __restrict__ dst,
    const float* __restrict__ Q, const float* __restrict__ KR,
    const float* __restrict__ pri, float* __restrict__ sc,
    float* __restrict__ mb, int E) {
  int idx = blockIdx.x * blockDim.x + threadIdx.x;
  if (idx >= E * NHEAD) return;
  int e = idx >> 3, h = idx & 7;
  int s = src[e], d = dst[e];
  const float4* qp = (const float4*)(Q + (size_t)d * 128 + h * 16);
  const float4* kp = (const float4*)(KR + (size_t)s * 128 + h * 16);
  float acc = 0.f;
#pragma unroll
  for (int j = 0; j < 4; ++j) {
    float4 a = qp[j], b = kp[j];
    acc += a.x * b.x + a.y * b.y + a.z * b.z + a.w * b.w;
  }
  float v = acc * pri[h] * 0.25f;  // 1/sqrt(16)
  sc[idx] = v;
  atomicMaxF(mb + (size_t)d * NHEAD + h, v);
}

// pass 2: ex = exp(score - max[dst]); seg-sum
__global__ __launch_bounds__(256) void edge_exp_sum(
    const int* __restrict__ dst, float* __restrict__ sc,
    const float* __restrict__ mb, float* __restrict__ den, int E) {
  int idx = blockIdx.x * blockDim.x + threadIdx.x;
  if (idx >= E * NHEAD) return;
  int e = idx >> 3, h = idx & 7;
  int d = dst[e];
  float ex = __expf(sc[idx] - mb[(size_t)d * NHEAD + h]);
  sc[idx] = ex;
  atomicAdd(den + (size_t)d * NHEAD + h, ex);
}

// pass 3: attn = ex/den[dst]; agg[dst] += vr[src]*attn
__global__ __launch_bounds__(256) void edge_scatter(
    const int* __restrict__ src, const int* __restrict__ dst,
    const float* __restrict__ sc, const float* __restrict__ den,
    const float* __restrict__ VR, float* __restrict__ agg, int E) {
  int idx = blockIdx.x * blockDim.x + threadIdx.x;
  if (idx >= E * NHEAD) return;
  int e = idx >> 3, h = idx & 7;
  int s = src[e], d = dst[e];
  float attn = sc[idx] / den[(size_t)d * NHEAD + h];
  const float4* vp = (const float4*)(VR + (size_t)s * 128 + h * 16);
  float* ap = agg + (size_t)d * 128 + h * 16;
#pragma unroll
  for (int j = 0; j < 4; ++j) {
    float4 v = vp[j];
    atomicAdd(ap + j * 4 + 0, v.x * attn);
    atomicAdd(ap + j * 4 + 1, v.y * attn);
    atomicAdd(ap + j * 4 + 2, v.z * attn);
    atomicAdd(ap + j * 4 + 3, v.w * attn);
  }
}

__global__ __launch_bounds__(256) void fill_f32(float* __restrict__ p, float v,
                                                size_t n) {
  size_t i = (size_t)blockIdx.x * blockDim.x + threadIdx.x;
  if (i < n) p[i] = v;
}

__global__ __launch_bounds__(256) void mean2(const float* __restrict__ a,
                                             const float* __restrict__ b,
                                             float* __restrict__ o, size_t n) {
  size_t i = (size_t)blockIdx.x * blockDim.x + threadIdx.x;
  if (i < n) o[i] = 0.5f * (a[i] + b[i]);
}

// ---------------------------------------------------------------------------
// Host orchestration
// ---------------------------------------------------------------------------
extern "C" void kernel_launch(void* const* d_in, const int* in_sizes, int n_in,
                              void* d_out, int out_size, void* d_ws,
                              size_t ws_size, hipStream_t stream) {
  const float* h0 = (const float*)d_in[0];
  const float* h1 = (const float*)d_in[1];
  const int* src0 = (const int*)d_in[2];
  const int* dst0 = (const int*)d_in[3];
  const int* src1 = (const int*)d_in[4];
  const int* dst1 = (const int*)d_in[5];
  const int* src2 = (const int*)d_in[6];
  const int* dst2 = (const int*)d_in[7];
  const float* Wk = (const float*)d_in[8];
  const float* bk = (const float*)d_in[9];
  const float* Wq = (const float*)d_in[10];
  const float* bq = (const float*)d_in[11];
  const float* Wv = (const float*)d_in[12];
  const float* bv = (const float*)d_in[13];
  const float* Wa = (const float*)d_in[14];
  const float* ba = (const float*)d_in[15];
  const float* rel_att = (const float*)d_in[16];
  const float* rel_msg = (const float*)d_in[17];
  const float* rel_pri = (const float*)d_in[18];
  const float* skip = (const float*)d_in[19];

  const int N0 = in_sizes[0] / 128;
  const int N1 = in_sizes[1] / 128;
  const int E = in_sizes[2];

  // ---- workspace carve-out (fp32) ----
  float* ws = (float*)d_ws;
  size_t off = 0;
  auto alloc = [&](size_t n) {
    float* p = ws + off;
    off += n;
    return p;
  };
  const size_t n0f = (size_t)N0 * 128, n1f = (size_t)N1 * 128;
  float* Q0 = alloc(n0f);
  float* Q1 = alloc(n1f);
  float* KR0 = alloc(n0f);  // rel0: src type 0
  float* KR1 = alloc(n1f);  // rel1: src type 1
  float* KR2 = alloc(n0f);  // rel2: src type 0
  float* VR0 = alloc(n0f);
  float* VR1 = alloc(n1f);
  float* VR2 = alloc(n0f);
  float* AGG0 = alloc(n1f);  // rel0 -> dst type 1
  float* AGG1 = alloc(n0f);  // rel1 -> dst type 0
  float* AGG2 = alloc(n0f);  // rel2 -> dst type 0
  float* TME0 = alloc(n0f);
  float* SC0 = alloc((size_t)E * 8);
  float* SC1 = alloc((size_t)E * 8);
  float* SC2 = alloc((size_t)E * 8);
  float* MB0 = alloc((size_t)N1 * 8);
  float* MB1 = alloc((size_t)N0 * 8);
  float* MB2 = alloc((size_t)N0 * 8);
  float* DN0 = alloc((size_t)N1 * 8);
  float* DN1 = alloc((size_t)N0 * 8);
  float* DN2 = alloc((size_t)N0 * 8);
  float* WF = alloc(6 * 16384);  // fused weights: KA0,KA1,KA2,VM0,VM1,VM2
  float* BF = alloc(6 * 128);

  const int WMAT = 16384;  // 128*128
  const int REL = 2048;    // 8*16*16

  // ---- 1) fuse relation transforms into projection weights ----
  dim3 fb(64), ft(256);
  fuse_weights<<<fb, ft, 0, stream>>>(Wk + 0 * WMAT, bk + 0, rel_att + 0 * REL,
                                      WF + 0 * WMAT, BF + 0 * 128);
  fuse_weights<<<fb, ft, 0, stream>>>(Wk + 1 * WMAT, bk + 128,
                                      rel_att + 1 * REL, WF + 1 * WMAT,
                                      BF + 1 * 128);
  fuse_weights<<<fb, ft, 0, stream>>>(Wk + 0 * WMAT, bk + 0, rel_att + 2 * REL,
                                      WF + 2 * WMAT, BF + 2 * 128);
  fuse_weights<<<fb, ft, 0, stream>>>(Wv + 0 * WMAT, bv + 0, rel_msg + 0 * REL,
                                      WF + 3 * WMAT, BF + 3 * 128);
  fuse_weights<<<fb, ft, 0, stream>>>(Wv + 1 * WMAT, bv + 128,
                                      rel_msg + 1 * REL, WF + 4 * WMAT,
                                      BF + 4 * 128);
  fuse_weights<<<fb, ft, 0, stream>>>(Wv + 0 * WMAT, bv + 0, rel_msg + 2 * REL,
                                      WF + 5 * WMAT, BF + 5 * 128);

  // ---- 2) projection GEMMs (fp32 WMMA) ----
  dim3 gt(256);
  dim3 gb0((N0 + 127) / 128), gb1((N1 + 127) / 128);
  gemm128_wmma<<<gb0, gt, 0, stream>>>(h0, Wq + 0 * WMAT, bq + 0, Q0, nullptr,
                                       nullptr, N0);
  gemm128_wmma<<<gb1, gt, 0, stream>>>(h1, Wq + 1 * WMAT, bq + 128, Q1, nullptr,
                                       nullptr, N1);
  gemm128_wmma<<<gb0, gt, 0, stream>>>(h0, WF + 0 * WMAT, BF + 0 * 128, KR0,
                                       nullptr, nullptr, N0);
  gemm128_wmma<<<gb1, gt, 0, stream>>>(h1, WF + 1 * WMAT, BF + 1 * 128, KR1,
                                       nullptr, nullptr, N1);
  gemm128_wmma<<<gb0, gt, 0, stream>>>(h0, WF + 2 * WMAT, BF + 2 * 128, KR2,
                                       nullptr, nullptr, N0);
  gemm128_wmma<<<gb0, gt, 0, stream>>>(h0, WF + 3 * WMAT, BF + 3 * 128, VR0,
                                       nullptr, nullptr, N0);
  gemm128_wmma<<<gb1, gt, 0, stream>>>(h1, WF + 4 * WMAT, BF + 4 * 128, VR1,
                                       nullptr, nullptr, N1);
  gemm128_wmma<<<gb0, gt, 0, stream>>>(h0, WF + 5 * WMAT, BF + 5 * 128, VR2,
                                       nullptr, nullptr, N0);

  // ---- 3) re-init accumulators (graph-replay safe) ----
  auto fill = [&](float* p, float v, size_t n) {
    fill_f32<<<dim3((unsigned)((n + 255) / 256)), dim3(256), 0, stream>>>(p, v,
                                                                          n);
  };
  fill(AGG0, 0.f, n1f);
  fill(AGG1, 0.f, n0f);
  fill(AGG2, 0.f, n0f);
  fill(DN0, 0.f, (size_t)N1 * 8);
  fill(DN1, 0.f, (size_t)N0 * 8);
  fill(DN2, 0.f, (size_t)N0 * 8);
  fill(MB0, -__builtin_inff(), (size_t)N1 * 8);
  fill(MB1, -__builtin_inff(), (size_t)N0 * 8);
  fill(MB2, -__builtin_inff(), (size_t)N0 * 8);

  // ---- 4) edge softmax + scatter per relation ----
  dim3 eb((unsigned)(((size_t)E * 8 + 255) / 256)), et(256);
  // rel 0: type0 -> type1
  edge_score_max<<<eb, et, 0, stream>>>(src0, dst0, Q1, KR0, rel_pri + 0, SC0,
                                        MB0, E);
  edge_exp_sum<<<eb, et, 0, stream>>>(dst0, SC0, MB0, DN0, E);
  edge_scatter<<<eb, et, 0, stream>>>(src0, dst0, SC0, DN0, VR0, AGG0, E);
  // rel 1: type1 -> type0
  edge_score_max<<<eb, et, 0, stream>>>(src1, dst1, Q0, KR1, rel_pri + 8, SC1,
                                        MB1, E);
  edge_exp_sum<<<eb, et, 0, stream>>>(dst1, SC1, MB1, DN1, E);
  edge_scatter<<<eb, et, 0, stream>>>(src1, dst1, SC1, DN1, VR1, AGG1, E);
  // rel 2: type0 -> type0
  edge_score_max<<<eb, et, 0, stream>>>(src2, dst2, Q0, KR2, rel_pri + 16, SC2,
                                        MB2, E);
  edge_exp_sum<<<eb, et, 0, stream>>>(dst2, SC2, MB2, DN2, E);
  edge_scatter<<<eb, et, 0, stream>>>(src2, dst2, SC2, DN2, VR2, AGG2, E);

  // ---- 5) cross-relation mean for type 0 (type 1 has a single relation) ----
  mean2<<<dim3((unsigned)((n0f + 255) / 256)), dim3(256), 0, stream>>>(
      AGG1, AGG2, TME0, n0f);

  // ---- 6) output GEMMs with sigmoid-skip residual epilogue ----
  float* out = (float*)d_out;
  gemm128_wmma<<<gb0, gt, 0, stream>>>(TME0, Wa + 0 * WMAT, ba + 0, out, h0,
                                       skip + 0, N0);
  gemm128_wmma<<<gb1, gt, 0, stream>>>(AGG0, Wa + 1 * WMAT, ba + 128,
                                       out + n0f, h1, skip + 1, N1);
}